// Resize_38104949850218
// MI455X (gfx1250) — compile-verified
//
#include <hip/hip_runtime.h>
#include <stdint.h>

// ---------------------------------------------------------------------------
// Trilinear resize [4,128,128,128,2] -> [4,192,192,192,2], zoom = 1.5/axis.
// Memory-bound (~292-320 MB total traffic => ~13us floor @ 23.3 TB/s).
// Strategy: per-block 12x12x12 output tile; TDM (tensor_load_to_lds) stages
// the 9x9x9(x2ch) input box into LDS in one DMA; threads gather 8 corners
// from LDS and emit 8-byte non-temporal stores.
// ---------------------------------------------------------------------------

typedef __attribute__((ext_vector_type(4))) unsigned int u32x4;
typedef __attribute__((ext_vector_type(8))) int          i32x8;
typedef __attribute__((ext_vector_type(4))) int          i32x4;
typedef __attribute__((ext_vector_type(2))) float        f32x2;

#define IN_N   128
#define OUT_N  192
#define TILE_O 12      // output tile edge (multiple of 3 so input aligns)
#define TILE_I 9       // max input extent per axis (12 * 2/3 + 1)
#define NTILE  16      // 192 / 12
#define BLK    256

__device__ __forceinline__ int imin(int a, int b) { return a < b ? a : b; }

// Reference-faithful per-axis linear weights:
//   loc  = o / 1.5 ; cloc = clip(loc, 0, 127)
//   i0   = clip(floor(loc), 0, 127) ; i1 = min(i0+1, 127)
//   w0   = i1 - cloc ; w1 = 1 - w0
__device__ __forceinline__ void axis_w(int o, int ibase,
                                       float& w0, int& j0, int& j1) {
  float loc = (float)o / 1.5f;            // loc >= 0 always
  float cl  = fminf(loc, 127.0f);
  int i0 = imin((int)loc, 127);
  int i1 = imin(i0 + 1, 127);
  w0 = (float)i1 - cl;
  j0 = i0 - ibase;
  j1 = i1 - ibase;
}

__global__ __launch_bounds__(BLK) void resize_trilinear_tdm(
    const float* __restrict__ vol, float* __restrict__ out) {
  // 9*9*18 floats = 5832 B of LDS
  __shared__ __align__(16) float tile[TILE_I * TILE_I * TILE_I * 2];

  const int tileId = blockIdx.x;              // 0..4095
  const int b  = blockIdx.y;                  // 0..3
  const int tx = tileId & 15;
  const int ty = (tileId >> 4) & 15;
  const int tz = tileId >> 8;

  const int ox0 = tx * TILE_O, oy0 = ty * TILE_O, oz0 = tz * TILE_O;
  const int ix0 = tx * 8,      iy0 = ty * 8,      iz0 = tz * 8;   // 12*2/3 = 8
  const int ex  = imin(TILE_I, IN_N - ix0);   // 9, or 8 at the edge
  const int ey  = imin(TILE_I, IN_N - iy0);
  const int ez  = imin(TILE_I, IN_N - iz0);
  const int ex2 = ex * 2;                     // LDS row pitch in floats (x*c)

  // ---- Tensor Data Mover: stage the input box into LDS (wave 0 only) ----
  if (threadIdx.x < 32) {
    // Byte address of tile origin: (((b*128+iz0)*128+iy0)*128+ix0)*2 floats
    const uint64_t ga =
        (uint64_t)(uintptr_t)vol +
        8ull * (uint64_t)((( (long long)b * IN_N + iz0) * IN_N + iy0) * IN_N + ix0);
    // Flat shared-aperture address: low 32 bits == LDS byte offset (ISA 10.2)
    const uint32_t lds = (uint32_t)(uintptr_t)&tile[0];

    // D# group 0: count=1 | lds_addr | global_addr[56:0] | type=2
    u32x4 g0 = { 1u,
                 lds,
                 (uint32_t)ga,
                 (uint32_t)((ga >> 32) & 0x01FFFFFFu) | (2u << 30) };

    // D# group 1:
    //  w0: workgroup_mask=0, data_size=2 (4B) at bits 17:16
    //  w1[31:16] = tensor_dim0[15:0] = 256 (one x*c row, in elements)
    //  w2[31:16] = tensor_dim1[15:0] = 128 (y)
    //  w3[31:16] = tile_dim0 = ex*2 ; w4 = tile_dim1 | tile_dim2<<16
    //  w5 = tensor_dim0_stride = 256 ; w6[31:16] = tensor_dim1_stride[15:0]=32768
    i32x8 g1 = { (2 << 16),
                 (256 << 16),
                 (128 << 16),
                 (ex2 << 16),
                 (ey | (ez << 16)),
                 256,
                 (int)0x80000000u,      // 32768 << 16
                 0 };

    // D# group 2: tensor_dim2=128 ; tensor_dim2_stride = 128*128*256 elements
    i32x4 g2 = { 128, 0, 4194304, 0 };
    i32x4 g3 = { 0, 0, 0, 0 };
    // 6-arg toolchain variant: extra i32x8 group (unused here) before cpol
    i32x8 g4 = { 0, 0, 0, 0, 0, 0, 0, 0 };

    __builtin_amdgcn_tensor_load_to_lds(g0, g1, g2, g3, g4, 0);
    __builtin_amdgcn_s_wait_tensorcnt(0);   // TENSORcnt -> 0 before barrier
  }
  __syncthreads();

  // ---- Gather-interpolate from LDS, one float2 (both channels) per point ----
  const long long obase =
      (((long long)(b * OUT_N + oz0) * OUT_N + oy0) * OUT_N + ox0);

  for (int p = threadIdx.x; p < TILE_O * TILE_O * TILE_O; p += BLK) {
    const int lx  = p % TILE_O;
    const int rem = p / TILE_O;
    const int ly  = rem % TILE_O;
    const int lz  = rem / TILE_O;

    float wx0, wy0, wz0; int jx0, jx1, jy0, jy1, jz0, jz1;
    axis_w(ox0 + lx, ix0, wx0, jx0, jx1);
    axis_w(oy0 + ly, iy0, wy0, jy0, jy1);
    axis_w(oz0 + lz, iz0, wz0, jz0, jz1);
    const float wx1 = 1.0f - wx0, wy1 = 1.0f - wy0, wz1 = 1.0f - wz0;

    const int zy00 = (jz0 * ey + jy0) * ex2;
    const int zy01 = (jz0 * ey + jy1) * ex2;
    const int zy10 = (jz1 * ey + jy0) * ex2;
    const int zy11 = (jz1 * ey + jy1) * ex2;
    const int x0 = jx0 * 2, x1 = jx1 * 2;     // even -> 8B aligned ds_load_b64

    const f32x2 v000 = *(const f32x2*)&tile[zy00 + x0];
    const f32x2 v001 = *(const f32x2*)&tile[zy00 + x1];
    const f32x2 v010 = *(const f32x2*)&tile[zy01 + x0];
    const f32x2 v011 = *(const f32x2*)&tile[zy01 + x1];
    const f32x2 v100 = *(const f32x2*)&tile[zy10 + x0];
    const f32x2 v101 = *(const f32x2*)&tile[zy10 + x1];
    const f32x2 v110 = *(const f32x2*)&tile[zy11 + x0];
    const f32x2 v111 = *(const f32x2*)&tile[zy11 + x1];

    const float c00 = wz0 * wy0, c01 = wz0 * wy1;
    const float c10 = wz1 * wy0, c11 = wz1 * wy1;
    const float w000 = c00 * wx0, w001 = c00 * wx1;
    const float w010 = c01 * wx0, w011 = c01 * wx1;
    const float w100 = c10 * wx0, w101 = c10 * wx1;
    const float w110 = c11 * wx0, w111 = c11 * wx1;

    f32x2 r;
    r.x = v000.x * w000 + v001.x * w001 + v010.x * w010 + v011.x * w011 +
          v100.x * w100 + v101.x * w101 + v110.x * w110 + v111.x * w111;
    r.y = v000.y * w000 + v001.y * w001 + v010.y * w010 + v011.y * w011 +
          v100.y * w100 + v101.y * w101 + v110.y * w110 + v111.y * w111;

    const long long oidx = obase + ((long long)lz * OUT_N + ly) * OUT_N + lx;
    // Write-once streaming output: non-temporal, keep L2 for the input set.
    __builtin_nontemporal_store(r, (f32x2*)(out + 2 * oidx));
  }
}

extern "C" void kernel_launch(void* const* d_in, const int* in_sizes, int n_in,
                              void* d_out, int out_size, void* d_ws, size_t ws_size,
                              hipStream_t stream) {
  const float* vol = (const float*)d_in[0];
  float* out = (float*)d_out;
  (void)in_sizes; (void)n_in; (void)out_size; (void)d_ws; (void)ws_size;

  dim3 grid(NTILE * NTILE * NTILE, 4);   // 4096 tiles x 4 batches
  resize_trilinear_tdm<<<grid, BLK, 0, stream>>>(vol, out);
}